// GATMILModel_5454608466149
// MI455X (gfx1250) — compile-verified
//
#include <hip/hip_runtime.h>
#include <hip/hip_bf16.h>
#include <math.h>

// ---------------------------------------------------------------------------
// GAT-MIL fused per-batch kernel for MI455X (gfx1250, wave32, WMMA f32)
// B=2048, N=110, layer1: 5 heads x 24 (concat->120), layer2: 3 heads x 3 (avg)
// One workgroup (256 thr = 8 wave32) per batch element; everything in LDS.
// B-operand matrices are stored K-pair-interleaved so each WMMA B-fragment
// loads as a single ds_load_b64 (2 LDS issues per v_wmma instead of 3).
// ---------------------------------------------------------------------------

typedef float v2f __attribute__((ext_vector_type(2)));
typedef float v8f __attribute__((ext_vector_type(8)));

#define NN 110
#define NP 112            // 7 * 16 row/col padding for 16x16 tiles
#define XM_LD 116         // Xm / attn LDS row stride (bank-conflict padded)
#define F1_LD 128         // feats1 col count (120 padded)
#define H1_LD 132         // h1 stride (padded, non-64-multiple for A-frag reads)
#define WB1_LD 128
#define F2_LD 32
#define C1 120
#define GAT_ALPHA 0.2f
#define NEGBIG (-9.0e15f)

// interleaved pair-row strides (words)
#define F1_PLD  (2*F1_LD)   // 256
#define WB1_PLD (2*WB1_LD)  // 256
#define WB2_PLD 32
#define F2_PLD  (2*F2_LD)   // 64

// LDS word offsets (floats)
#define OFF_XM   0
#define OFF_F1   (OFF_XM + NP*XM_LD)          // 12992
#define OFF_H1   (OFF_F1 + NP*F1_LD)          // +14336
#define OFF_WB1  (OFF_H1 + NP*H1_LD)          // +14784
#define OFF_WB2  (OFF_WB1 + NP*WB1_LD)        // +14336
#define OFF_F2   (OFF_WB2 + 128*16)           // +2048
#define OFF_ES1  (OFF_F2 + NP*F2_LD)          // +3584
#define OFF_EN1  (OFF_ES1 + 5*NP)
#define OFF_ES2  (OFF_EN1 + 5*NP)
#define OFF_EN2  (OFF_ES2 + 3*NP)
#define OFF_ABIT (OFF_EN2 + 3*NP)             // 112*4 u32 adjacency bitmask
#define OFF_H2   (OFF_ABIT + NP*4)
#define OFF_T    (OFF_H2 + NP*4)
#define OFF_S    (OFF_T + NP)
#define SMEM_WORDS (OFF_S + NP)               // 64992 words = 259,968 B < 320 KB

__device__ __forceinline__ float lrelu(float x) { return x > 0.f ? x : GAT_ALPHA * x; }

// One 16x16 output tile, K-loop over WMMA_F32_16X16X4_F32.
// Ap: pre-offset to A row (m = lane&15), row-major, contiguous in K (8B aligned).
// Bp: pre-offset to B col (n = lane&15) in pair-interleaved layout; the
//     (k,k+1) pair is contiguous -> one ds_load_b64 per fragment.
__device__ __forceinline__ v8f wmma_tile(const float* Ap, const float* Bp,
                                         int ldbp, int K, int koff, v8f acc) {
  for (int k0 = 0; k0 < K; k0 += 4) {
    v2f a = *(const v2f*)(Ap + k0 + koff);
    v2f b = *(const v2f*)(Bp + ((k0 + koff) >> 1) * ldbp);
    acc = __builtin_amdgcn_wmma_f32_16x16x4_f32(false, a, false, b,
                                                (short)0, acc, false, false);
  }
  return acc;
}

// Masked leaky-relu attention logits + row softmax, written into the
// (reused) OFF_XM buffer. One row per wave; 32 lanes cover 110 columns.
__device__ __forceinline__ void attn_rows(float* smem, const float* es,
                                          const float* en, const unsigned* abit,
                                          int lane, int wv) {
  for (int n = wv; n < NN; n += 8) {
    float esv = es[n];
    float vals[4];
    float mx = -3.0e38f;
#pragma unroll
    for (int j = 0; j < 4; ++j) {
      int m2 = lane + 32 * j;
      float v = -3.0e38f;
      if (m2 < NN) {
        float e = lrelu(esv + en[m2]);
        unsigned bit = (abit[n * 4 + (m2 >> 5)] >> (m2 & 31)) & 1u;
        v = bit ? e : NEGBIG;
      }
      vals[j] = v;
      mx = fmaxf(mx, v);
    }
#pragma unroll
    for (int off = 16; off > 0; off >>= 1) mx = fmaxf(mx, __shfl_xor(mx, off, 32));
    float s = 0.f;
#pragma unroll
    for (int j = 0; j < 4; ++j) {
      int m2 = lane + 32 * j;
      float e = 0.f;
      if (m2 < NN) e = __expf(vals[j] - mx);
      vals[j] = e;
      s += e;
    }
#pragma unroll
    for (int off = 16; off > 0; off >>= 1) s += __shfl_xor(s, off, 32);
    float inv = 1.f / s;
#pragma unroll
    for (int j = 0; j < 4; ++j) {
      int m2 = lane + 32 * j;
      if (m2 < XM_LD) smem[OFF_XM + n * XM_LD + m2] = (m2 < NN) ? vals[j] * inv : 0.f;
    }
  }
}

// Precompute sigmoid(M) and zero-padded, K-pair-interleaved weight matrices.
extern "C" __global__ void gat_prep(const float* __restrict__ M,
                                    const float* __restrict__ W1,
                                    const float* __restrict__ W2,
                                    float* __restrict__ sigM,
                                    float* __restrict__ Wb1,
                                    float* __restrict__ Wb2) {
  int i = blockIdx.x * blockDim.x + threadIdx.x;
  if (i < NN * NN) sigM[i] = 1.f / (1.f + __expf(-M[i]));
  if (i < NP * WB1_LD) {                       // logical [K=110 pad 112][120 pad 128]
    int k = i / WB1_LD, c = i % WB1_LD;
    float v = 0.f;
    if (k < NN && c < C1) { int h = c / 24, f = c % 24; v = W1[(h * NN + k) * 24 + f]; }
    Wb1[(k >> 1) * WB1_PLD + c * 2 + (k & 1)] = v;
  }
  if (i < 128 * 16) {                          // logical [K=120 pad 128][9 pad 16]
    int k = i / 16, c = i % 16;
    float v = 0.f;
    if (k < C1 && c < 9) { int h = c / 3, f = c % 3; v = W2[(h * C1 + k) * 3 + f]; }
    Wb2[(k >> 1) * WB2_PLD + c * 2 + (k & 1)] = v;
  }
}

extern "C" __global__ __launch_bounds__(256)
void gat_mil_kernel(const float* __restrict__ X, const float* __restrict__ A,
                    const float* __restrict__ as1, const float* __restrict__ an1,
                    const float* __restrict__ b1,
                    const float* __restrict__ as2, const float* __restrict__ an2,
                    const float* __restrict__ b2,
                    const float* __restrict__ fc1, const float* __restrict__ fc2,
                    const float* __restrict__ sigM, const float* __restrict__ Wb1,
                    const float* __restrict__ Wb2,
                    float* __restrict__ out) {
  extern __shared__ float smem[];
  unsigned* abit = (unsigned*)(smem + OFF_ABIT);
  const int tid = threadIdx.x;
  const int lane = tid & 31;
  const int wv = tid >> 5;
  const int b = blockIdx.x;
  const float* Xb = X + (size_t)b * NN * NN;
  const float* Ag = A + (size_t)b * NN * NN;

  __builtin_prefetch(Xb + tid * 47, 0, 0);     // global_prefetch_b8

  // zero init (padding rows/cols must be 0 for WMMA K-loops)
  for (int i = tid; i < NP * XM_LD; i += 256) smem[OFF_XM + i] = 0.f;
  for (int i = tid; i < NP * F2_LD; i += 256) smem[OFF_F2 + i] = 0.f;
  for (int i = tid; i < NP * 4; i += 256) abit[i] = 0u;
  // stage weights in LDS (already interleaved in global scratch)
  for (int i = tid; i < NP * WB1_LD; i += 256) smem[OFF_WB1 + i] = Wb1[i];
  for (int i = tid; i < 128 * 16; i += 256) smem[OFF_WB2 + i] = Wb2[i];
  __syncthreads();

  // gated input Xm = X * sigmoid(M), adjacency bitmask
  for (int i = tid; i < NN * NN; i += 256) {
    int r = i / NN, c = i % NN;
    smem[OFF_XM + r * XM_LD + c] = Xb[i] * sigM[i];
    if (Ag[i] > 0.f) atomicOr(&abit[r * 4 + (c >> 5)], 1u << (c & 31));
  }
  __syncthreads();

  const int m = lane & 15;
  const int koff = (lane >> 4) << 1;   // A/B frag K offset for upper half-wave
  const int rb = (lane >> 4) << 3;     // C/D frag row offset for upper half-wave

  // ---- layer 1 projection: feats1 = Xm[112x112] @ Wb1[112x128] ----
  // feats1 stored pair-interleaved (it is the B operand of the aggregation).
  for (int mt = 0; mt < 7; ++mt) {
    v8f acc = {0, 0, 0, 0, 0, 0, 0, 0};
    const float* Ap = smem + OFF_XM + (mt * 16 + m) * XM_LD;
    const float* Bp = smem + OFF_WB1 + (wv * 16 + m) * 2;
    acc = wmma_tile(Ap, Bp, WB1_PLD, NP, koff, acc);
#pragma unroll
    for (int v = 0; v < 8; ++v) {
      int row = mt * 16 + rb + v;
      smem[OFF_F1 + (row >> 1) * F1_PLD + (wv * 16 + m) * 2 + (row & 1)] = acc[v];
    }
  }
  __syncthreads();

  // per-head attention logit vectors e_s, e_n
  for (int p = tid; p < 5 * NN; p += 256) {
    int h = p / NN, n = p % NN;
    float s1 = 0.f, s2 = 0.f;
    for (int f = 0; f < 24; ++f) {
      float fv = smem[OFF_F1 + (n >> 1) * F1_PLD + (h * 24 + f) * 2 + (n & 1)];
      s1 += fv * as1[h * 24 + f];
      s2 += fv * an1[h * 24 + f];
    }
    smem[OFF_ES1 + h * NP + n] = s1;
    smem[OFF_EN1 + h * NP + n] = s2;
  }
  __syncthreads();

  // ---- layer 1 heads: softmax(attn) then h1 = attn @ feats1_head + b1 ----
  for (int h = 0; h < 5; ++h) {
    attn_rows(smem, smem + OFF_ES1 + h * NP, smem + OFF_EN1 + h * NP, abit, lane, wv);
    __syncthreads();
    for (int t = wv; t < 14; t += 8) {         // 7 row tiles x 2 col tiles
      int mt = t % 7, ct = t / 7;
      v8f acc = {0, 0, 0, 0, 0, 0, 0, 0};
      const float* Ap = smem + OFF_XM + (mt * 16 + m) * XM_LD;
      const float* Bp = smem + OFF_F1 + (h * 24 + ct * 16 + m) * 2;
      acc = wmma_tile(Ap, Bp, F1_PLD, NP, koff, acc);
      int f = ct * 16 + m;
      if (f < 24) {
        float bias = b1[f];
#pragma unroll
        for (int v = 0; v < 8; ++v) {
          int row = mt * 16 + rb + v;
          float val = 0.f;
          if (row < NN) val = lrelu(acc[v] + bias);
          smem[OFF_H1 + row * H1_LD + h * 24 + f] = val;   // row-major (A operand)
        }
      }
    }
    __syncthreads();
  }

  // ---- layer 2 projection: feats2 = h1[112x120] @ Wb2[120x16] ----
  if (wv < 7) {
    v8f acc = {0, 0, 0, 0, 0, 0, 0, 0};
    const float* Ap = smem + OFF_H1 + (wv * 16 + m) * H1_LD;
    const float* Bp = smem + OFF_WB2 + m * 2;
    acc = wmma_tile(Ap, Bp, WB2_PLD, C1, koff, acc);
#pragma unroll
    for (int v = 0; v < 8; ++v) {
      int row = wv * 16 + rb + v;
      smem[OFF_F2 + (row >> 1) * F2_PLD + m * 2 + (row & 1)] = acc[v];
    }
  }
  __syncthreads();

  for (int p = tid; p < 3 * NN; p += 256) {
    int h = p / NN, n = p % NN;
    float s1 = 0.f, s2 = 0.f;
    for (int f = 0; f < 3; ++f) {
      float fv = smem[OFF_F2 + (n >> 1) * F2_PLD + (h * 3 + f) * 2 + (n & 1)];
      s1 += fv * as2[h * 3 + f];
      s2 += fv * an2[h * 3 + f];
    }
    smem[OFF_ES2 + h * NP + n] = s1;
    smem[OFF_EN2 + h * NP + n] = s2;
  }
  __syncthreads();

  // ---- layer 2 heads, accumulated across heads in the WMMA C operand ----
  // B-frag col offset 3*h makes tile col j correspond to feature j each head.
  v8f acc2 = {0, 0, 0, 0, 0, 0, 0, 0};
  for (int h = 0; h < 3; ++h) {
    attn_rows(smem, smem + OFF_ES2 + h * NP, smem + OFF_EN2 + h * NP, abit, lane, wv);
    __syncthreads();
    if (wv < 7) {
      const float* Ap = smem + OFF_XM + (wv * 16 + m) * XM_LD;
      const float* Bp = smem + OFF_F2 + (h * 3 + m) * 2;
      acc2 = wmma_tile(Ap, Bp, F2_PLD, NP, koff, acc2);
    }
    __syncthreads();
  }
  if (wv < 7 && m < 3) {
    float bias = b2[m];
#pragma unroll
    for (int v = 0; v < 8; ++v) {
      int row = wv * 16 + rb + v;
      if (row < NN) smem[OFF_H2 + row * 4 + m] = lrelu(acc2[v] * (1.f / 3.f) + bias);
    }
  }
  __syncthreads();

  // ---- readout: t = sigmoid(h2 @ fc1); p = softmax(t @ fc2); out = sum t*p ----
  if (tid < NN) {
    float z = smem[OFF_H2 + tid * 4 + 0] * fc1[0]
            + smem[OFF_H2 + tid * 4 + 1] * fc1[1]
            + smem[OFF_H2 + tid * 4 + 2] * fc1[2];
    smem[OFF_T + tid] = 1.f / (1.f + __expf(-z));
  }
  __syncthreads();
  if (tid < NN) {
    float s = 0.f;
    for (int n = 0; n < NN; ++n) s += smem[OFF_T + n] * fc2[n * NN + tid];
    smem[OFF_S + tid] = s;
  }
  __syncthreads();
  if (tid == 0) {
    float mx = -3.0e38f;
    for (int n = 0; n < NN; ++n) mx = fmaxf(mx, smem[OFF_S + n]);
    float se = 0.f, st = 0.f;
    for (int n = 0; n < NN; ++n) {
      float e = __expf(smem[OFF_S + n] - mx);
      se += e;
      st += smem[OFF_T + n] * e;
    }
    out[b] = st / se;
  }
}

extern "C" void kernel_launch(void* const* d_in, const int* in_sizes, int n_in,
                              void* d_out, int out_size, void* d_ws, size_t ws_size,
                              hipStream_t stream) {
  const float* X   = (const float*)d_in[0];
  const float* A   = (const float*)d_in[1];
  const float* M   = (const float*)d_in[2];
  const float* W1  = (const float*)d_in[3];
  const float* as1 = (const float*)d_in[4];
  const float* an1 = (const float*)d_in[5];
  const float* b1  = (const float*)d_in[6];
  const float* W2  = (const float*)d_in[7];
  const float* as2 = (const float*)d_in[8];
  const float* an2 = (const float*)d_in[9];
  const float* b2  = (const float*)d_in[10];
  const float* fc1 = (const float*)d_in[11];
  const float* fc2 = (const float*)d_in[12];
  float* out = (float*)d_out;

  float* sigM = (float*)d_ws;            // 12100 floats
  float* Wb1  = sigM + NN * NN;          // 14336 floats (interleaved)
  float* Wb2  = Wb1 + NP * WB1_LD;       // 2048 floats (interleaved)

  int Bn = in_sizes[0] / (NN * NN);      // 2048

  gat_prep<<<(NP * WB1_LD + 255) / 256, 256, 0, stream>>>(M, W1, W2, sigM, Wb1, Wb2);

  size_t shmem = (size_t)SMEM_WORDS * sizeof(float);   // ~260 KB (<320 KB/WGP)
  gat_mil_kernel<<<Bn, 256, shmem, stream>>>(X, A, as1, an1, b1, as2, an2, b2,
                                             fc1, fc2, sigM, Wb1, Wb2, out);
}